// HyperGraphConv_54391465837246
// MI455X (gfx1250) — compile-verified
//
#include <hip/hip_runtime.h>
#include <hip/hip_bf16.h>
#include <math.h>

#define BB 8
#define NN 4096
#define DD 256
#define KK 8
#define SPAD 132   // 128 cols + 4 pad (bank-conflict-free)
#define MPAD 260   // 256 + 4 pad

typedef __attribute__((ext_vector_type(2))) float v2f;
typedef __attribute__((ext_vector_type(8))) float v8f;

static __device__ __forceinline__ v8f wmma_f32(v2f a, v2f b, v8f c) {
  // V_WMMA_F32_16X16X4_F32 : D = A(16x4) * B(4x16) + C(16x16), fp32 throughout
  return __builtin_amdgcn_wmma_f32_16x16x4_f32(false, a, false, b, (short)0, c,
                                               false, false);
}

// ---------------------------------------------------------------------------
// Kernel 1: L2-normalize rows of X, store transposed AND pair-interleaved:
//   XT2[b][d>>1][n] = float2( Xn[n][2*(d>>1)], Xn[n][2*(d>>1)+1] )
// so a WMMA fragment (k,k+1 per lane) is ONE coalesced b64 load.
// ---------------------------------------------------------------------------
__global__ void __launch_bounds__(256) k_norm_t(const float* __restrict__ X,
                                                float* __restrict__ XT) {
  const int wave = threadIdx.x >> 5, lane = threadIdx.x & 31;
  const int row = blockIdx.x * 8 + wave;          // 0 .. B*N-1
  const float* xr = X + (size_t)row * DD;
  float v[8];
  float s = 0.f;
#pragma unroll
  for (int i = 0; i < 8; ++i) {
    v[i] = xr[lane + 32 * i];
    s += v[i] * v[i];
  }
#pragma unroll
  for (int m = 1; m < 32; m <<= 1) s += __shfl_xor(s, m, 32);
  const float scale = 1.f / fmaxf(sqrtf(s), 1e-12f);
  const int b = row >> 12;           // row / N
  const int n = row & (NN - 1);      // row % N
  float* xt = XT + (size_t)b * DD * NN;
#pragma unroll
  for (int i = 0; i < 8; ++i) {
    const int d = lane + 32 * i;
    xt[(size_t)(d >> 1) * (2 * NN) + 2 * n + (d & 1)] = v[i] * scale;
  }
}

// ---------------------------------------------------------------------------
// Kernel 2: W (E,D) -> Wt2 pair-interleaved: Wt2[d>>1][e] = (W[e][2p],W[e][2p+1])
// ---------------------------------------------------------------------------
__global__ void __launch_bounds__(256) k_wt(const float* __restrict__ W,
                                            float* __restrict__ Wt) {
  const int t = blockIdx.x * 256 + threadIdx.x;   // < 65536
  const int e = t >> 8, d = t & 255;
  Wt[(size_t)(d >> 1) * (2 * DD) + 2 * e + (d & 1)] = W[(size_t)e * DD + d];
}

// ---------------------------------------------------------------------------
// Kernel 3: sim = Xn @ Xn^T (fp32 WMMA, pair-interleaved operands) fused with
// running top-8. Workgroup = 8 waves = one 16-row panel; 32 chunks of 128 cols.
// Inner loop: 1 b64 load : 1 WMMA, two independent accumulation chains.
// ---------------------------------------------------------------------------
__global__ void __launch_bounds__(256) k_sim_topk(const float* __restrict__ XTf,
                                                  int* __restrict__ idx) {
  __shared__ float simL[16 * SPAD];
  const int b  = blockIdx.x >> 8;                 // N/16 = 256 tiles / batch
  const int r0 = (blockIdx.x & 255) << 4;
  const int wave = threadIdx.x >> 5, lane = threadIdx.x & 31;
  const int half = lane >> 4, lm = lane & 15;
  const v2f* Xb = (const v2f*)(XTf + (size_t)b * DD * NN);  // [128][NN] pairs

  // A fragments (16-row panel) hoisted into registers: one b64 load each.
  // A 16x4 layout: lanes 0-15 hold K={k0,k0+1}, lanes 16-31 K={k0+2,k0+3};
  // pair index = 2j + half.
  v2f afrag[64];
#pragma unroll
  for (int j = 0; j < 64; ++j)
    afrag[j] = Xb[(size_t)(2 * j + half) * NN + r0 + lm];

  // per-row top-8 state (threads 0..15 own one row each)
  float tv[KK];
  int   ti[KK];
#pragma unroll
  for (int p = 0; p < KK; ++p) { tv[p] = -3.4e38f; ti[p] = 0; }

  for (int chunk = 0; chunk < 32; ++chunk) {
    const int c0 = chunk * 128 + wave * 16;
    v8f accA = {0.f, 0.f, 0.f, 0.f, 0.f, 0.f, 0.f, 0.f};
    v8f accB = {0.f, 0.f, 0.f, 0.f, 0.f, 0.f, 0.f, 0.f};
#pragma unroll
    for (int j = 0; j < 64; j += 2) {
      const v2f b0 = Xb[(size_t)(2 * j + half) * NN + c0 + lm];
      const v2f b1 = Xb[(size_t)(2 * j + 2 + half) * NN + c0 + lm];
      accA = wmma_f32(afrag[j],     b0, accA);
      accB = wmma_f32(afrag[j + 1], b1, accB);
    }
    // D layout: lane holds col n=lm, rows m = v + 8*half
#pragma unroll
    for (int v = 0; v < 8; ++v)
      simL[(v + 8 * half) * SPAD + wave * 16 + lm] = accA[v] + accB[v];
    __syncthreads();

    if (threadIdx.x < 16) {
      const int r = threadIdx.x;
      for (int j = 0; j < 128; ++j) {
        float val = simL[r * SPAD + j];
        if (val > tv[KK - 1]) {              // fast reject of most candidates
          int gi = chunk * 128 + j;
          // stable bubble-insert (strict > : earlier index wins ties, as jax)
#pragma unroll
          for (int p = 0; p < KK; ++p) {
            const bool gt = val > tv[p];
            const float keep  = gt ? val : tv[p];
            const int   keepi = gt ? gi  : ti[p];
            const float pass  = gt ? tv[p] : val;
            const int   passi = gt ? ti[p] : gi;
            tv[p] = keep; ti[p] = keepi;
            val = pass; gi = passi;
          }
        }
      }
    }
    __syncthreads();
  }

  if (threadIdx.x < 16) {
    int* o = idx + ((size_t)b * NN + r0 + threadIdx.x) * KK;
#pragma unroll
    for (int p = 0; p < KK; ++p) o[p] = ti[p];
  }
}

// ---------------------------------------------------------------------------
// Kernel 4: gather+mean -> msg (LDS), Y = msg @ W^T + bias (fp32 WMMA),
// residual add, LayerNorm — fully fused per 16-row panel.
// ---------------------------------------------------------------------------
__global__ void __launch_bounds__(256) k_msg_ln(
    const float* __restrict__ X, const float* __restrict__ Wtf,
    const float* __restrict__ bias, const float* __restrict__ gamma,
    const float* __restrict__ beta, const int* __restrict__ idx,
    float* __restrict__ out) {
  __shared__ float msgL[16 * MPAD];
  __shared__ float zL[16 * MPAD];
  const int b  = blockIdx.x >> 8;
  const int r0 = (blockIdx.x & 255) << 4;
  const int wave = threadIdx.x >> 5, lane = threadIdx.x & 31;
  const int half = lane >> 4, lm = lane & 15;
  const float* Xb = X + (size_t)b * NN * DD;
  const v2f* Wt2 = (const v2f*)Wtf;               // [128][DD] pairs

  // ---- gather + mean: 16 threads per row, coalesced 64B segments ----
  {
    const int r  = threadIdx.x >> 4;
    const int dl = threadIdx.x & 15;
    const int* ri = idx + ((size_t)b * NN + r0 + r) * KK;
    float accm[16];
#pragma unroll
    for (int i = 0; i < 16; ++i) accm[i] = 0.f;
    for (int k = 0; k < KK; ++k) {
      const float* xr = Xb + (size_t)ri[k] * DD;
#pragma unroll
      for (int i = 0; i < 16; ++i) accm[i] += xr[16 * i + dl];
    }
#pragma unroll
    for (int i = 0; i < 16; ++i)
      msgL[r * MPAD + 16 * i + dl] = accm[i] * 0.125f;
  }
  __syncthreads();

  // ---- GEMM: each wave produces two 16x16 output tiles (e0, e0+16) ----
  {
    const int e0 = wave * 32, e1 = wave * 32 + 16;
    v8f acc0 = {0.f, 0.f, 0.f, 0.f, 0.f, 0.f, 0.f, 0.f};
    v8f acc1 = {0.f, 0.f, 0.f, 0.f, 0.f, 0.f, 0.f, 0.f};
#pragma unroll
    for (int j = 0; j < 64; ++j) {
      const int k = 4 * j + 2 * half;               // even; 8B-aligned in LDS
      const v2f a  = *(const v2f*)&msgL[lm * MPAD + k];   // one ds_load_b64
      const v2f b0 = Wt2[(size_t)(2 * j + half) * DD + e0 + lm];
      const v2f b1 = Wt2[(size_t)(2 * j + half) * DD + e1 + lm];
      acc0 = wmma_f32(a, b0, acc0);
      acc1 = wmma_f32(a, b1, acc1);
    }
#pragma unroll
    for (int v = 0; v < 8; ++v) {
      const int m = v + 8 * half;
      const float* xrow = Xb + (size_t)(r0 + m) * DD;
      zL[m * MPAD + e0 + lm] = xrow[e0 + lm] + acc0[v] + bias[e0 + lm];
      zL[m * MPAD + e1 + lm] = xrow[e1 + lm] + acc1[v] + bias[e1 + lm];
    }
  }
  __syncthreads();

  // ---- LayerNorm: each wave handles rows {2w, 2w+1} via half-wave shfl ----
  {
    const int r = wave * 2 + half;
    float s = 0.f, s2 = 0.f;
#pragma unroll
    for (int i = 0; i < 16; ++i) {
      const float z = zL[r * MPAD + lm + 16 * i];
      s += z; s2 += z * z;
    }
#pragma unroll
    for (int m = 1; m < 16; m <<= 1) {   // masks 1,2,4,8 stay in 16-lane half
      s  += __shfl_xor(s,  m, 32);
      s2 += __shfl_xor(s2, m, 32);
    }
    const float mu  = s * (1.f / 256.f);
    const float var = s2 * (1.f / 256.f) - mu * mu;
    const float inv = rsqrtf(var + 1e-5f);
    float* orow = out + ((size_t)b * NN + r0 + r) * DD;
#pragma unroll
    for (int i = 0; i < 16; ++i) {
      const int d = lm + 16 * i;
      const float z = zL[r * MPAD + d];
      orow[d] = (z - mu) * inv * gamma[d] + beta[d];
    }
  }
}

// ---------------------------------------------------------------------------
extern "C" void kernel_launch(void* const* d_in, const int* in_sizes, int n_in,
                              void* d_out, int out_size, void* d_ws,
                              size_t ws_size, hipStream_t stream) {
  (void)in_sizes; (void)n_in; (void)out_size; (void)ws_size;
  const float* X     = (const float*)d_in[0];
  const float* W     = (const float*)d_in[1];
  const float* bias  = (const float*)d_in[2];
  const float* gamma = (const float*)d_in[3];
  const float* beta  = (const float*)d_in[4];
  float* out = (float*)d_out;

  // workspace layout
  float* XT = (float*)d_ws;                        // B*D*N floats (32 MB)
  float* Wt = XT + (size_t)BB * DD * NN;           // D*D floats (256 KB)
  int*  idx = (int*)(Wt + (size_t)DD * DD);        // B*N*K ints (1 MB)

  k_norm_t  <<<BB * NN / 8,    256, 0, stream>>>(X, XT);
  k_wt      <<<DD * DD / 256,  256, 0, stream>>>(W, Wt);
  k_sim_topk<<<BB * (NN / 16), 256, 0, stream>>>(XT, idx);
  k_msg_ln  <<<BB * (NN / 16), 256, 0, stream>>>(X, Wt, bias, gamma, beta, idx,
                                                 out);
}